// SubViewEncoder_32693291057233
// MI455X (gfx1250) — compile-verified
//
#include <hip/hip_runtime.h>

typedef __attribute__((ext_vector_type(2))) float v2f;
typedef __attribute__((ext_vector_type(8))) float v8f;

#define N_NODES 100000
#define NFEAT   128
#define NHID    64

// ---------------------------------------------------------------------------
// Degree / normalization kernels
// ---------------------------------------------------------------------------
__global__ __launch_bounds__(256) void deg_init_kernel(float* __restrict__ deg, int n) {
    int i = blockIdx.x * blockDim.x + threadIdx.x;
    if (i < n) deg[i] = 1.0f;  // self-loop contribution
}

__global__ __launch_bounds__(256) void deg_count_kernel(const int* __restrict__ dst,
                                                        float* __restrict__ deg, int n_edges) {
    int e = blockIdx.x * blockDim.x + threadIdx.x;
    if (e < n_edges) atomicAdd(&deg[dst[e]], 1.0f);
}

__global__ __launch_bounds__(256) void deg_rsqrt_kernel(float* __restrict__ deg, int n) {
    int i = blockIdx.x * blockDim.x + threadIdx.x;
    if (i < n) deg[i] = rsqrtf(deg[i]);  // in-place -> dinv
}

// ---------------------------------------------------------------------------
// Dense GEMM via V_WMMA_F32_16X16X4_F32 (full fp32 precision) with fused
// self-loop epilogue:
//   H[M,64]   = A[M,K] @ W[K,64]                  (needed for edge gather)
//   ACC[M,64] = H * dinv[row]^2                   (scatter accumulator init)
// Block = 128 threads = 4 waves; wave w handles N-tile w (16 cols),
// blockIdx.x handles one 16-row M-tile.  K % 4 == 0.
// A 16x4 layout: lanes 0-15 = M rows; half-wave selects K {0,1} vs {2,3}.
// B 4x16 layout: lanes 0-15 = N cols; half-wave selects K {0,1} vs {2,3}.
// D 16x16 layout: VGPR r -> row (r + 8*halfwave), col = lane&15.
// ---------------------------------------------------------------------------
__global__ __launch_bounds__(128) void gemm_wmma_f32(const float* __restrict__ A,
                                                     const float* __restrict__ W,
                                                     const float* __restrict__ dinv,
                                                     float* __restrict__ H,
                                                     float* __restrict__ ACC, int K) {
    const int N = NHID;
    const int lane = threadIdx.x & 31;
    const int wave = threadIdx.x >> 5;
    const int m0 = blockIdx.x << 4;
    const int n0 = wave << 4;
    const int ml = lane & 15;
    const int hi = lane >> 4;

    const float* arow = A + (size_t)(m0 + ml) * K + hi * 2;   // 8B aligned (K even)
    const float* bp   = W + (size_t)(hi * 2) * N + n0 + ml;

    v8f d = {};
    for (int k = 0; k < K; k += 4) {
        v2f a = *(const v2f*)(arow + k);      // A[m][k + hi*2 .. +1]
        v2f b;
        b.x = bp[(size_t)k * N];              // W[k + hi*2    ][n]
        b.y = bp[(size_t)k * N + N];          // W[k + hi*2 + 1][n]
        d = __builtin_amdgcn_wmma_f32_16x16x4_f32(
            false, a, false, b, (short)0, d, false, false);
    }

    const size_t base = (size_t)(m0 + hi * 8) * N + n0 + ml;
    float* hrow = H + base;
    float* crow = ACC + base;
#pragma unroll
    for (int r = 0; r < 8; ++r) {
        float v  = d[r];
        float di = dinv[m0 + hi * 8 + r];     // broadcast within half-wave (cached)
        hrow[(size_t)r * N] = v;
        crow[(size_t)r * N] = v * di * di;
    }
}

// ---------------------------------------------------------------------------
// Edge scatter: 16 lanes per edge, 4 channels each (coalesced 256B gather per
// edge, then float atomic adds into acc[dst]).
// ---------------------------------------------------------------------------
__global__ __launch_bounds__(256) void scatter_kernel(const float* __restrict__ h,
                                                      const int* __restrict__ src,
                                                      const int* __restrict__ dst,
                                                      const float* __restrict__ dinv,
                                                      float* __restrict__ acc, int n_edges) {
    long long tid = (long long)blockIdx.x * blockDim.x + threadIdx.x;
    int e = (int)(tid >> 4);
    int q = (int)(tid & 15);
    if (e >= n_edges) return;
    int s = src[e], d = dst[e];
    float norm = dinv[s] * dinv[d];
    const float4 v = *(const float4*)(h + (size_t)s * NHID + q * 4);
    float* p = acc + (size_t)d * NHID + q * 4;
    atomicAdd(p + 0, v.x * norm);
    atomicAdd(p + 1, v.y * norm);
    atomicAdd(p + 2, v.z * norm);
    atomicAdd(p + 3, v.w * norm);
}

// ---------------------------------------------------------------------------
// out = relu(acc + bias)                        (layer-1 finalize -> residual)
// out = relu(acc + bias) + res                  (layer-2 finalize -> d_out)
// ---------------------------------------------------------------------------
__global__ __launch_bounds__(256) void bias_relu_kernel(const float* __restrict__ acc,
                                                        const float* __restrict__ bias,
                                                        float* __restrict__ out) {
    int t = blockIdx.x * blockDim.x + threadIdx.x;
    if (t < N_NODES * NHID)
        out[t] = fmaxf(acc[t] + bias[t & (NHID - 1)], 0.0f);
}

__global__ __launch_bounds__(256) void bias_relu_res_kernel(const float* __restrict__ acc,
                                                            const float* __restrict__ bias,
                                                            const float* __restrict__ res,
                                                            float* __restrict__ out) {
    int t = blockIdx.x * blockDim.x + threadIdx.x;
    if (t < N_NODES * NHID)
        out[t] = fmaxf(acc[t] + bias[t & (NHID - 1)], 0.0f) + res[t];
}

// ---------------------------------------------------------------------------
extern "C" void kernel_launch(void* const* d_in, const int* in_sizes, int n_in,
                              void* d_out, int out_size, void* d_ws, size_t ws_size,
                              hipStream_t stream) {
    const float* x  = (const float*)d_in[0];
    const int*   ei = (const int*)d_in[1];
    const float* W1 = (const float*)d_in[2];
    const float* b1 = (const float*)d_in[3];
    const float* W2 = (const float*)d_in[4];
    const float* b2 = (const float*)d_in[5];

    const int n_edges = in_sizes[1] / 2;
    const int* src = ei;
    const int* dst = ei + n_edges;

    const size_t NH = (size_t)N_NODES * NHID;   // 6,400,000 floats
    float* dinv = (float*)d_ws;                 // [N]
    float* h0   = dinv + N_NODES;               // [N,64] gemm out / reused as h2
    float* h1   = h0 + NH;                      // [N,64] residual
    float* acc  = h1 + NH;                      // [N,64] scatter accumulator

    const unsigned gN   = (N_NODES + 255) / 256;
    const unsigned gE   = (unsigned)((n_edges + 255) / 256);
    const unsigned gNH  = (unsigned)((NH + 255) / 256);
    const unsigned gSc  = (unsigned)(((long long)n_edges * 16 + 255) / 256);
    const unsigned gGem = N_NODES / 16;         // 6250 M-tiles

    // ---- normalization ----
    deg_init_kernel<<<gN, 256, 0, stream>>>(dinv, N_NODES);
    deg_count_kernel<<<gE, 256, 0, stream>>>(dst, dinv, n_edges);
    deg_rsqrt_kernel<<<gN, 256, 0, stream>>>(dinv, N_NODES);

    // ---- layer 1 (GEMM writes h0 and self-loop-initialized acc) ----
    gemm_wmma_f32<<<gGem, 128, 0, stream>>>(x, W1, dinv, h0, acc, NFEAT);
    scatter_kernel<<<gSc, 256, 0, stream>>>(h0, src, dst, dinv, acc, n_edges);
    bias_relu_kernel<<<gNH, 256, 0, stream>>>(acc, b1, h1);

    // ---- layer 2 (h0 reused as h2) ----
    gemm_wmma_f32<<<gGem, 128, 0, stream>>>(h1, W2, dinv, h0, acc, NHID);
    scatter_kernel<<<gSc, 256, 0, stream>>>(h0, src, dst, dinv, acc, n_edges);
    bias_relu_res_kernel<<<gNH, 256, 0, stream>>>(acc, b2, h1, (float*)d_out);
}